// DisplacementLayer_89747636617238
// MI455X (gfx1250) — compile-verified
//
#include <hip/hip_runtime.h>
#include <cstdint>
#include <cstddef>

typedef float f4 __attribute__((ext_vector_type(4)));
typedef float f2 __attribute__((ext_vector_type(2)));

namespace {
constexpr int kB = 8, kH = 512, kW = 512, kC = 16;
constexpr int kCq = kC / 4;            // float4 quads per pixel
constexpr unsigned kLogW = 9;          // W = 512
constexpr unsigned kLogHW = 18;        // H*W = 2^18
}

// One lane per (pixel, channel-quad): 4 lanes cooperate on one pixel.
// wave32: 8 pixels per wave, output stores fully contiguous b128.
__global__ __launch_bounds__(256) void warp_bilinear_kernel(
    const float* __restrict__ x_t, const float* __restrict__ uv,
    float* __restrict__ out, unsigned npix) {
  unsigned gid = blockIdx.x * blockDim.x + threadIdx.x;
  unsigned p = gid >> 2;   // pixel index
  unsigned q = gid & 3u;   // channel quad (0..3)
  if (p >= npix) return;

  unsigned wx = p & (kW - 1);
  unsigned hy = (p >> kLogW) & (kH - 1);
  unsigned b  = p >> kLogHW;

  // uv read-once -> non-temporal load (th:NT); 4 lanes of a pixel hit the
  // same 8B, merged per-wave by the TA.
  f2 d = __builtin_nontemporal_load(reinterpret_cast<const f2*>(uv) + p);
  float xq = (float)wx + d.x;
  float yq = (float)hy + d.y;

  // tfa semantics: floor clamped to [0, size-2], alpha clamped to [0,1]
  float fx = fminf(fmaxf(floorf(xq), 0.0f), (float)(kW - 2));
  float fy = fminf(fmaxf(floorf(yq), 0.0f), (float)(kH - 2));
  float ax = fminf(fmaxf(xq - fx, 0.0f), 1.0f);
  float ay = fminf(fmaxf(yq - fy, 0.0f), 1.0f);
  int ix = (int)fx;
  int iy = (int)fy;

  // Base address of this lane's channel quad at (b, iy, ix).
  const f4* src = reinterpret_cast<const f4*>(x_t)
                + (size_t)(((b * kH) + iy) * kW + (unsigned)ix) * kCq + q;

  // gfx1250 prefetch of the bottom gather row (global_prefetch_b8).
  __builtin_prefetch(src + (size_t)kW * kCq, 0, 3);

  // 4-corner gather, 16B each; tl/tr (and bl/br) are 64B apart, and the
  // row stride (32KB) fits the 24-bit immediate offset -> 4 loads off one base.
  f4 tl = src[0];
  f4 tr = src[kCq];
  f4 bl = src[(size_t)kW * kCq];
  f4 br = src[(size_t)kW * kCq + kCq];

  f4 top = tl + (tr - tl) * ax;
  f4 bot = bl + (br - bl) * ax;
  f4 res = top + (bot - top) * ay;

  // Output is write-once streaming (128 MiB) -> non-temporal b128 store.
  __builtin_nontemporal_store(res, reinterpret_cast<f4*>(out) + (size_t)p * kCq + q);
}

extern "C" void kernel_launch(void* const* d_in, const int* in_sizes, int n_in,
                              void* d_out, int out_size, void* d_ws, size_t ws_size,
                              hipStream_t stream) {
  const float* x_t = (const float*)d_in[0];  // (B,H,W,C) float32
  const float* uv  = (const float*)d_in[1];  // (B,H,W,2) float32
  float* out = (float*)d_out;                // (B,H,W,C) float32

  unsigned npix = (unsigned)(in_sizes[1] / 2);   // B*H*W
  unsigned nthreads = npix * 4u;                 // 4 lanes per pixel
  dim3 block(256);
  dim3 grid((nthreads + block.x - 1) / block.x);
  warp_bilinear_kernel<<<grid, block, 0, stream>>>(x_t, uv, out, npix);
}